// HamiltonianPotential_82772609729196
// MI455X (gfx1250) — compile-verified
//
#include <hip/hip_runtime.h>
#include <hip/hip_bf16.h>
#include <math.h>

// ---------- types ----------
typedef __attribute__((ext_vector_type(2))) float v2f;
typedef __attribute__((ext_vector_type(8))) float v8f;

#define Bdim 4
#define Ndim 128
#define Kdim 32
#define KK   1024
#define Vdim 32000
#define EPSF 1e-8f
#define LAMBDA_BELIEF 1.0f

__device__ inline v8f wmma_f32(v2f a, v2f b, v8f c) {
#if defined(__HIP_DEVICE_COMPILE__)
  // V_WMMA_F32_16X16X4_F32 : D = A(16x4) * B(4x16) + C(16x16), all fp32
  return __builtin_amdgcn_wmma_f32_16x16x4_f32(false, a, false, b, (short)0, c,
                                               false, false);
#else
  (void)a; (void)b;
  return c;
#endif
}

// ---------------------------------------------------------------------------
// Kernel 1: per-node precompute.  One block (256 threads) per node m = b*N+n.
//   R      = expm(sum_a phi_a G_a)            (Taylor, generators antisym)
//   Lq     = chol(Sigma + eps I), ld = 2*sum log diag
//   Y      = Lq^{-1} R ;  A = Y^T Y = R^T (Sigma+eps)^{-1} R
//   C      = R^T (Sigma + eps I + mu mu^T) R
//   u      = R^T mu ; w = A u ; c = u.w
//   selfkl = 0.5*(tr(Sp^{-1}(Sq+eps)) + d^T Sp^{-1} d - 32 + ldp - ldq)
// ---------------------------------------------------------------------------
__device__ void cholesky32(float* S, float* L, int tid) {
  for (int k = 0; k < 32; ++k) {
    if (tid == 0) L[k * 32 + k] = sqrtf(S[k * 32 + k]);
    __syncthreads();
    if (tid < 32 && tid > k) L[tid * 32 + k] = S[tid * 32 + k] / L[k * 32 + k];
    __syncthreads();
    for (int p = 0; p < 4; ++p) {
      int e = tid + p * 256;
      int i = e >> 5, j = e & 31;
      if (i > k && j > k) S[e] -= L[i * 32 + k] * L[j * 32 + k];
    }
    __syncthreads();
  }
}

__global__ __launch_bounds__(256) void node_precompute(
    const float* __restrict__ mu, const float* __restrict__ Sigma,
    const float* __restrict__ phi, const float* __restrict__ mu_prior,
    const float* __restrict__ Sigma_prior, const float* __restrict__ gen,
    float* __restrict__ Aflat, float* __restrict__ Cflat,
    float* __restrict__ uOut, float* __restrict__ wOut,
    float* __restrict__ cjOut, float* __restrict__ ldOut,
    float* __restrict__ selfklOut) {
  __shared__ float sM[KK], sR[KK], sB1[KK], sB2[KK], sB3[KK], sB4[KK], sL[KK];
  __shared__ float sMu[32], sU[32], sW[32], sVec[32], sRed[32];
  const int m = blockIdx.x;
  const int tid = threadIdx.x;
  const float p0 = phi[m * 3 + 0], p1 = phi[m * 3 + 1], p2 = phi[m * 3 + 2];

  // --- stage 1: M = sum phi_a * G_a ---
  for (int p = 0; p < 4; ++p) {
    int e = tid + p * 256;
    sM[e] = p0 * gen[e] + p1 * gen[KK + e] + p2 * gen[2 * KK + e];
  }
  __syncthreads();

  // --- stage 2: R = expm(M) via 12-term Taylor (||M|| << 1) ---
  for (int p = 0; p < 4; ++p) {
    int e = tid + p * 256;
    int i = e >> 5, j = e & 31;
    float v = sM[e];
    sR[e] = v + (i == j ? 1.0f : 0.0f);
    sB1[e] = v;
  }
  __syncthreads();
  {
    float* cur = sB1;
    float* nxt = sB2;
    for (int t = 2; t <= 12; ++t) {
      float inv = 1.0f / (float)t;
      float vals[4];
      for (int p = 0; p < 4; ++p) {
        int e = tid + p * 256;
        int i = e >> 5, j = e & 31;
        float acc = 0.f;
        for (int k = 0; k < 32; ++k) acc += cur[i * 32 + k] * sM[k * 32 + j];
        vals[p] = acc * inv;
      }
      for (int p = 0; p < 4; ++p) {
        int e = tid + p * 256;
        nxt[e] = vals[p];
        sR[e] += vals[p];
      }
      __syncthreads();
      float* tsw = cur; cur = nxt; nxt = tsw;
    }
  }

  // --- stage 3: Cholesky of Sigma+eps -> sL ; ldq ---
  for (int p = 0; p < 4; ++p) {
    int e = tid + p * 256;
    int i = e >> 5, j = e & 31;
    sB3[e] = Sigma[(size_t)m * KK + e] + (i == j ? EPSF : 0.f);
    sL[e] = 0.f;
  }
  __syncthreads();
  cholesky32(sB3, sL, tid);
  if (tid < 32) sRed[tid] = logf(sL[tid * 32 + tid]);
  __syncthreads();
  float ldq = 0.f;
  if (tid == 0) {
    float a = 0.f;
    for (int i = 0; i < 32; ++i) a += sRed[i];
    ldq = 2.f * a;
    ldOut[m] = ldq;
  }
  __syncthreads();

  // --- stage 4: forward solve Lq Y = R -> sB1 ; A = Y^T Y -> sM + global ---
  for (int i = 0; i < 32; ++i) {
    if (tid < 32) {
      float acc = sR[i * 32 + tid];
      for (int k = 0; k < i; ++k) acc -= sL[i * 32 + k] * sB1[k * 32 + tid];
      sB1[i * 32 + tid] = acc / sL[i * 32 + i];
    }
    __syncthreads();
  }
  for (int p = 0; p < 4; ++p) {
    int e = tid + p * 256;
    int i = e >> 5, j = e & 31;
    float acc = 0.f;
    for (int k = 0; k < 32; ++k) acc += sB1[k * 32 + i] * sB1[k * 32 + j];
    sM[e] = acc;  // A (phiM no longer needed)
    Aflat[(size_t)m * KK + e] = acc;
  }
  __syncthreads();

  // --- stage 5: u = R^T mu ---
  if (tid < 32) sMu[tid] = mu[(size_t)m * 32 + tid];
  __syncthreads();
  if (tid < 32) {
    float acc = 0.f;
    for (int k = 0; k < 32; ++k) acc += sR[k * 32 + tid] * sMu[k];
    sU[tid] = acc;
    uOut[(size_t)m * 32 + tid] = acc;
  }
  __syncthreads();

  // --- stage 6: C = R^T (Sigma + eps I + mu mu^T) R ---
  for (int p = 0; p < 4; ++p) {
    int e = tid + p * 256;
    int i = e >> 5, j = e & 31;
    sB3[e] = Sigma[(size_t)m * KK + e] + sMu[i] * sMu[j] + (i == j ? EPSF : 0.f);
  }
  __syncthreads();
  for (int p = 0; p < 4; ++p) {
    int e = tid + p * 256;
    int i = e >> 5, j = e & 31;
    float acc = 0.f;
    for (int k = 0; k < 32; ++k) acc += sB3[i * 32 + k] * sR[k * 32 + j];
    sB2[e] = acc;
  }
  __syncthreads();
  for (int p = 0; p < 4; ++p) {
    int e = tid + p * 256;
    int i = e >> 5, j = e & 31;
    float acc = 0.f;
    for (int k = 0; k < 32; ++k) acc += sR[k * 32 + i] * sB2[k * 32 + j];
    Cflat[(size_t)m * KK + e] = acc;
  }
  __syncthreads();

  // --- stage 7: w = A u ; c = u.w ---
  if (tid < 32) {
    float acc = 0.f;
    for (int k = 0; k < 32; ++k) acc += sM[tid * 32 + k] * sU[k];
    sW[tid] = acc;
    wOut[(size_t)m * 32 + tid] = acc;
    sRed[tid] = acc * sU[tid];
  }
  __syncthreads();
  if (tid == 0) {
    float a = 0.f;
    for (int i = 0; i < 32; ++i) a += sRed[i];
    cjOut[m] = a;
  }
  __syncthreads();

  // --- stage 8: self-KL against prior ---
  for (int p = 0; p < 4; ++p) {
    int e = tid + p * 256;
    int i = e >> 5, j = e & 31;
    sB3[e] = Sigma_prior[(size_t)m * KK + e] + (i == j ? EPSF : 0.f);
    sL[e] = 0.f;
  }
  __syncthreads();
  cholesky32(sB3, sL, tid);
  if (tid < 32) sRed[tid] = logf(sL[tid * 32 + tid]);
  __syncthreads();
  float ldp = 0.f;
  if (tid == 0) {
    float a = 0.f;
    for (int i = 0; i < 32; ++i) a += sRed[i];
    ldp = 2.f * a;
  }
  __syncthreads();
  // RHS = Sigma_q + eps
  for (int p = 0; p < 4; ++p) {
    int e = tid + p * 256;
    int i = e >> 5, j = e & 31;
    sB2[e] = Sigma[(size_t)m * KK + e] + (i == j ? EPSF : 0.f);
  }
  __syncthreads();
  // Z = Lp^{-1} RHS -> sB1
  for (int i = 0; i < 32; ++i) {
    if (tid < 32) {
      float acc = sB2[i * 32 + tid];
      for (int k = 0; k < i; ++k) acc -= sL[i * 32 + k] * sB1[k * 32 + tid];
      sB1[i * 32 + tid] = acc / sL[i * 32 + i];
    }
    __syncthreads();
  }
  // X = Lp^{-T} Z -> sB4 ; trace = sum diag X
  for (int i = 31; i >= 0; --i) {
    if (tid < 32) {
      float acc = sB1[i * 32 + tid];
      for (int k = i + 1; k < 32; ++k) acc -= sL[k * 32 + i] * sB4[k * 32 + tid];
      sB4[i * 32 + tid] = acc / sL[i * 32 + i];
    }
    __syncthreads();
  }
  if (tid < 32) sRed[tid] = sB4[tid * 32 + tid];
  __syncthreads();
  if (tid < 32) sVec[tid] = mu_prior[(size_t)m * 32 + tid] - sMu[tid];
  __syncthreads();
  if (tid == 0) {
    float tr = 0.f;
    for (int i = 0; i < 32; ++i) tr += sRed[i];
    // y = Lp^{-1} d (serial, reuse sRed as y); mahal = |y|^2
    float mah = 0.f;
    for (int i = 0; i < 32; ++i) {
      float acc = sVec[i];
      for (int k = 0; k < i; ++k) acc -= sL[i * 32 + k] * sRed[k];
      float y = acc / sL[i * 32 + i];
      sRed[i] = y;
      mah += y * y;
    }
    selfklOut[m] = 0.5f * (tr + mah - 32.0f + ldp - ldq);  // ALPHA = 1
  }
}

// ---------------------------------------------------------------------------
// Kernel 2: pairwise alignment via WMMA f32 16x16x4.
// D[i,j] = <C_i, A_j>_F - 2 u_i.w_j  over K' = 1024 + 32.
// One wave per 16x16 output tile; 264 v_wmma per tile.
// ---------------------------------------------------------------------------
__global__ __launch_bounds__(32) void pair_align(
    const float* __restrict__ Aflat, const float* __restrict__ Cflat,
    const float* __restrict__ u, const float* __restrict__ w,
    const float* __restrict__ cj, const float* __restrict__ ld,
    const float* __restrict__ beta, float* __restrict__ alignPart) {
  const int tile = blockIdx.x;       // b*64 + ti*8 + tj
  const int b = tile >> 6;
  const int ti = (tile >> 3) & 7;
  const int tj = tile & 7;
  const int lane = threadIdx.x;
  const int half = lane >> 4;        // K sub-offset (ISA A/B frag layout)
  const int lr = lane & 15;
  const int iRow = b * Ndim + ti * 16 + lr;  // A-operand row (M)
  const int jRow = b * Ndim + tj * 16 + lr;  // B-operand col (N)
  const float* cp = Cflat + (size_t)iRow * KK + half * 2;
  const float* ap = Aflat + (size_t)jRow * KK + half * 2;
  v8f acc = {0.f, 0.f, 0.f, 0.f, 0.f, 0.f, 0.f, 0.f};
  for (int k0 = 0; k0 < KK; k0 += 4) {
    v2f av = *(const v2f*)(cp + k0);
    v2f bv = *(const v2f*)(ap + k0);
    acc = wmma_f32(av, bv, acc);
  }
  const float* up = u + (size_t)iRow * 32 + half * 2;
  const float* wp = w + (size_t)jRow * 32 + half * 2;
  for (int k0 = 0; k0 < 32; k0 += 4) {
    v2f av; av.x = -2.0f * up[k0]; av.y = -2.0f * up[k0 + 1];
    v2f bv; bv.x = wp[k0];         bv.y = wp[k0 + 1];
    acc = wmma_f32(av, bv, acc);
  }
  // epilogue: kl_ij = 0.5*(D + c_j - 32 + ld_j - ld_i); weight by beta, skip diag
  const int j = tj * 16 + lr;
  const int jn = b * Ndim + j;
  const float cjv = cj[jn];
  const float ldj = ld[jn];
  float ssum = 0.f;
#pragma unroll
  for (int r = 0; r < 8; ++r) {
    int i = ti * 16 + (half ? 8 + r : r);   // D-frag row mapping
    float ldi = ld[b * Ndim + i];
    float kl = 0.5f * (acc[r] + cjv - 32.0f + ldj - ldi);
    if (i != j) ssum += beta[((size_t)(b * Ndim + i)) * Ndim + j] * kl;
  }
  for (int off = 1; off < 32; off <<= 1) ssum += __shfl_xor(ssum, off, 32);
  if (lane == 0) alignPart[tile] = ssum;
}

// ---------------------------------------------------------------------------
// Kernel 3: cross-entropy via WMMA GEMM fused with online log-sum-exp.
// 32 blocks x 4 waves; block = 16 rows of mu, wave sweeps V/4 = 8000 cols.
// ---------------------------------------------------------------------------
__global__ __launch_bounds__(128) void ce_kernel(
    const float* __restrict__ mu, const float* __restrict__ Wout,
    const int* __restrict__ targets, float* __restrict__ cePart) {
  __shared__ float lm[4][16], ls[4][16], lt[4][16], lce[16];
  const int bb = blockIdx.x;  // 0..31 : b = bb>>3, rowBase = (bb&7)*16
  const int b = bb >> 3;
  const int rowBase = (bb & 7) * 16;
  const int wave = threadIdx.x >> 5;
  const int lane = threadIdx.x & 31;
  const int half = lane >> 4;
  const int lr = lane & 15;

  const float* murow = mu + (size_t)(b * Ndim + rowBase + lr) * 32 + half * 2;
  v2f afr[8];
#pragma unroll
  for (int kc = 0; kc < 8; ++kc) afr[kc] = *(const v2f*)(murow + kc * 4);
  int tgt[8];
#pragma unroll
  for (int r = 0; r < 8; ++r)
    tgt[r] = targets[b * Ndim + rowBase + (half ? 8 + r : r)];

  float m[8], s[8], tl[8];
#pragma unroll
  for (int r = 0; r < 8; ++r) { m[r] = -1e30f; s[r] = 0.f; tl[r] = 0.f; }

  const int vStart = wave * (Vdim / 4);
  const int vEnd = vStart + (Vdim / 4);
  for (int v0 = vStart; v0 < vEnd; v0 += 16) {
    const float* wrow = Wout + (size_t)(v0 + lr) * 32 + half * 2;
    v8f acc = {0.f, 0.f, 0.f, 0.f, 0.f, 0.f, 0.f, 0.f};
#pragma unroll
    for (int kc = 0; kc < 8; ++kc) {
      v2f bv = *(const v2f*)(wrow + kc * 4);
      acc = wmma_f32(afr[kc], bv, acc);
    }
    const int vcol = v0 + lr;
#pragma unroll
    for (int r = 0; r < 8; ++r) {
      float logit = acc[r];
      float M = fmaxf(m[r], logit);
      s[r] = s[r] * expf(m[r] - M) + expf(logit - M);
      m[r] = M;
      if (vcol == tgt[r]) tl[r] = logit;  // exactly one (lane,tile) hits
    }
  }
  // reduce across the 16 lanes that share a row (xor stays within half-wave)
#pragma unroll
  for (int off = 1; off < 16; off <<= 1) {
#pragma unroll
    for (int r = 0; r < 8; ++r) {
      float om = __shfl_xor(m[r], off, 32);
      float os = __shfl_xor(s[r], off, 32);
      float ot = __shfl_xor(tl[r], off, 32);
      float M = fmaxf(m[r], om);
      s[r] = s[r] * expf(m[r] - M) + os * expf(om - M);
      m[r] = M;
      tl[r] += ot;  // others contribute exact 0
    }
  }
  if (lr == 0) {
#pragma unroll
    for (int r = 0; r < 8; ++r) {
      int row = half ? 8 + r : r;
      lm[wave][row] = m[r];
      ls[wave][row] = s[r];
      lt[wave][row] = tl[r];
    }
  }
  __syncthreads();
  if (threadIdx.x < 16) {
    int row = threadIdx.x;
    float M = -1e30f, S = 0.f, T = 0.f;
    for (int wv = 0; wv < 4; ++wv) {
      float om = lm[wv][row], os = ls[wv][row];
      float Mn = fmaxf(M, om);
      S = S * expf(M - Mn) + os * expf(om - Mn);
      M = Mn;
      T += lt[wv][row];
    }
    lce[row] = (M + logf(S)) - T;  // logsumexp - target logit
  }
  __syncthreads();
  if (threadIdx.x == 0) {
    float a = 0.f;
    for (int r = 0; r < 16; ++r) a += lce[r];
    cePart[bb] = a;
  }
}

// ---------------------------------------------------------------------------
// Kernel 4: deterministic fixed-order final reduction into out[b].
// ---------------------------------------------------------------------------
__global__ __launch_bounds__(32) void final_reduce(
    const float* __restrict__ selfkl, const float* __restrict__ alignPart,
    const float* __restrict__ cePart, float* __restrict__ out) {
  int b = threadIdx.x;
  if (b < Bdim) {
    float acc = 0.f;
    for (int n = 0; n < Ndim; ++n) acc += selfkl[b * Ndim + n];
    float al = 0.f;
    for (int t = 0; t < 64; ++t) al += alignPart[b * 64 + t];
    acc += LAMBDA_BELIEF * al;
    float ce = 0.f;
    for (int wv = 0; wv < 8; ++wv) ce += cePart[b * 8 + wv];
    acc += ce;
    out[b] = acc;
  }
}

// ---------------------------------------------------------------------------
extern "C" void kernel_launch(void* const* d_in, const int* in_sizes, int n_in,
                              void* d_out, int out_size, void* d_ws,
                              size_t ws_size, hipStream_t stream) {
  (void)in_sizes; (void)n_in; (void)out_size; (void)ws_size;
  const float* mu          = (const float*)d_in[0];
  const float* Sigma       = (const float*)d_in[1];
  const float* phi         = (const float*)d_in[2];
  const float* mu_prior    = (const float*)d_in[3];
  const float* Sigma_prior = (const float*)d_in[4];
  const float* beta        = (const float*)d_in[5];
  const int*   targets     = (const int*)d_in[6];
  const float* Wout        = (const float*)d_in[7];
  const float* gen         = (const float*)d_in[8];
  float* out = (float*)d_out;

  // workspace layout (floats)
  float* ws        = (float*)d_ws;
  float* Aflat     = ws;                       // 512*1024
  float* Cflat     = Aflat + 512 * KK;         // 512*1024
  float* uArr      = Cflat + 512 * KK;         // 512*32
  float* wArr      = uArr + 512 * 32;          // 512*32
  float* cjArr     = wArr + 512 * 32;          // 512
  float* ldArr     = cjArr + 512;              // 512
  float* selfkl    = ldArr + 512;              // 512
  float* alignPart = selfkl + 512;             // 256
  float* cePart    = alignPart + 256;          // 32

  node_precompute<<<Bdim * Ndim, 256, 0, stream>>>(
      mu, Sigma, phi, mu_prior, Sigma_prior, gen, Aflat, Cflat, uArr, wArr,
      cjArr, ldArr, selfkl);
  pair_align<<<Bdim * 64, 32, 0, stream>>>(Aflat, Cflat, uArr, wArr, cjArr,
                                           ldArr, beta, alignPart);
  ce_kernel<<<32, 128, 0, stream>>>(mu, Wout, targets, cePart);
  final_reduce<<<1, 32, 0, stream>>>(selfkl, alignPart, cePart, out);
}